// GINLayer_49048526520633
// MI455X (gfx1250) — compile-verified
//
#include <hip/hip_runtime.h>

#define IN_DIM  128
#define HID     256
#define OUT_DIM 128
#define BN_EPS  1e-5f
#define LDA1    132   // padded LDS row stride for 16x128 A tile (conflict-free)
#define LDA2    260   // padded LDS row stride for 16x256 A tile

typedef __attribute__((ext_vector_type(2))) float v2f;
typedef __attribute__((ext_vector_type(8))) float v8f;

// ---------------------------------------------------------------- utilities
__global__ __launch_bounds__(256) void zero_kernel(float* __restrict__ p, size_t n) {
    size_t i = (size_t)blockIdx.x * blockDim.x + threadIdx.x;
    if (i < n) p[i] = 0.0f;
}

// ------------------------------------------------------- edge scatter (sum)
// one wave per edge: lane l handles features 4l..4l+3 (float4 load, 4 f32 atomics)
__global__ __launch_bounds__(256) void scatter_kernel(
    const float* __restrict__ x, const int* __restrict__ ei,
    float* __restrict__ aggr, int nedges)
{
    int lane  = threadIdx.x & 31;
    int wave  = (int)((blockIdx.x * blockDim.x + threadIdx.x) >> 5);
    int nwav  = (int)((gridDim.x * blockDim.x) >> 5);
    for (int e = wave; e < nedges; e += nwav) {
        int src = ei[e];            // row 0 of edge_index
        int dst = ei[nedges + e];   // row 1 of edge_index
        const float4 v = *(const float4*)(x + (size_t)src * IN_DIM + lane * 4);
        float* ap = aggr + (size_t)dst * IN_DIM + lane * 4;
        atomicAdd(ap + 0, v.x);
        atomicAdd(ap + 1, v.y);
        atomicAdd(ap + 2, v.z);
        atomicAdd(ap + 3, v.w);
    }
}

// ----------------------------------------------- GEMM1: h1 = A @ W1 + b1
// A = (1+eps)*x + aggr, staged in LDS. 8 waves/block, each a 16x16 N-tile.
__global__ __launch_bounds__(256) void gemm1_kernel(
    const float* __restrict__ x, const float* __restrict__ aggr,
    const float* __restrict__ eps_p, const float* __restrict__ W1,
    const float* __restrict__ b1, float* __restrict__ h1, int nrows)
{
    __shared__ float As[16 * LDA1];
    const int tid    = threadIdx.x;
    const int m_base = blockIdx.x * 16;
    const float e1   = 1.0f + eps_p[0];

    for (int i = tid; i < 16 * IN_DIM; i += 256) {
        int r = i >> 7, c = i & (IN_DIM - 1);
        int rr = m_base + r; if (rr >= nrows) rr = nrows - 1;
        size_t g = (size_t)rr * IN_DIM + c;
        As[r * LDA1 + c] = e1 * x[g] + aggr[g];
    }
    __syncthreads();

    const int wave   = tid >> 5;
    const int lane   = tid & 31;
    const int n_base = ((blockIdx.y << 3) + wave) << 4;   // 16 N-tiles over HID
    const int nc     = lane & 15;
    const int khalf  = lane >> 4;
    const float* Bp  = W1 + n_base + nc;

    v8f acc = {};
    for (int k0 = 0; k0 < IN_DIM; k0 += 4) {
        int ka = k0 + (khalf << 1);
        v2f a, b;
        a.x = As[nc * LDA1 + ka];
        a.y = As[nc * LDA1 + ka + 1];
        b.x = Bp[(size_t)ka * HID];
        b.y = Bp[(size_t)(ka + 1) * HID];
        acc = __builtin_amdgcn_wmma_f32_16x16x4_f32(
                  false, a, false, b, (short)0, acc, false, false);
    }
    float bias = b1[n_base + nc];
    float* outp = h1 + (size_t)m_base * HID + n_base + nc;
    #pragma unroll
    for (int v = 0; v < 8; ++v) {
        int m = v + (khalf << 3);
        if (m_base + m < nrows) outp[(size_t)m * HID] = acc[v] + bias;
    }
}

// --------------------------------------------- column sum / sumsq reduction
template<int C>
__global__ void colstats_kernel(const float* __restrict__ h, int nrows,
                                float* __restrict__ sum, float* __restrict__ sumsq)
{
    int c = threadIdx.x;                 // blockDim.x == C
    float s = 0.0f, q = 0.0f;
    for (int r = blockIdx.x; r < nrows; r += gridDim.x) {
        float v = h[(size_t)r * C + c];
        s += v; q += v * v;
    }
    atomicAdd(sum + c, s);
    atomicAdd(sumsq + c, q);
}

// BN -> affine form: y = x*scale + shift
__global__ void bnprep_kernel(const float* __restrict__ sum, const float* __restrict__ sumsq,
                              const float* __restrict__ gamma, const float* __restrict__ beta,
                              float* __restrict__ scale, float* __restrict__ shift,
                              int C, float invN)
{
    int c = blockIdx.x * blockDim.x + threadIdx.x;
    if (c < C) {
        float mu  = sum[c] * invN;
        float var = fmaxf(sumsq[c] * invN - mu * mu, 0.0f);
        float sc  = gamma[c] * rsqrtf(var + BN_EPS);
        scale[c] = sc;
        shift[c] = beta[c] - mu * sc;
    }
}

// --------------------------- GEMM2: out = relu(BN1(h1)) @ W2 + b2 (pre-BN2)
__global__ __launch_bounds__(256) void gemm2_kernel(
    const float* __restrict__ h1, const float* __restrict__ scale1,
    const float* __restrict__ shift1, const float* __restrict__ W2,
    const float* __restrict__ b2, float* __restrict__ out, int nrows)
{
    __shared__ float As[16 * LDA2];
    const int tid    = threadIdx.x;
    const int m_base = blockIdx.x * 16;

    for (int i = tid; i < 16 * HID; i += 256) {
        int r = i >> 8, c = i & (HID - 1);
        int rr = m_base + r; if (rr >= nrows) rr = nrows - 1;
        float v = h1[(size_t)rr * HID + c];
        As[r * LDA2 + c] = fmaxf(v * scale1[c] + shift1[c], 0.0f);
    }
    __syncthreads();

    const int wave   = tid >> 5;
    const int lane   = tid & 31;
    const int n_base = wave << 4;        // 8 N-tiles cover OUT_DIM
    const int nc     = lane & 15;
    const int khalf  = lane >> 4;
    const float* Bp  = W2 + n_base + nc;

    v8f acc = {};
    for (int k0 = 0; k0 < HID; k0 += 4) {
        int ka = k0 + (khalf << 1);
        v2f a, b;
        a.x = As[nc * LDA2 + ka];
        a.y = As[nc * LDA2 + ka + 1];
        b.x = Bp[(size_t)ka * OUT_DIM];
        b.y = Bp[(size_t)(ka + 1) * OUT_DIM];
        acc = __builtin_amdgcn_wmma_f32_16x16x4_f32(
                  false, a, false, b, (short)0, acc, false, false);
    }
    float bias = b2[n_base + nc];
    float* outp = out + (size_t)m_base * OUT_DIM + n_base + nc;
    #pragma unroll
    for (int v = 0; v < 8; ++v) {
        int m = v + (khalf << 3);
        if (m_base + m < nrows) outp[(size_t)m * OUT_DIM] = acc[v] + bias;
    }
}

// ------------------------------------ final BN2 + ReLU, in place on d_out
__global__ __launch_bounds__(256) void bnrelu_kernel(
    float* __restrict__ out, const float* __restrict__ scale,
    const float* __restrict__ shift, size_t n)
{
    size_t i = ((size_t)blockIdx.x * blockDim.x + threadIdx.x) * 4;
    if (i < n) {
        float4 v = *(float4*)(out + i);
        int c = (int)(i & (OUT_DIM - 1));
        v.x = fmaxf(v.x * scale[c + 0] + shift[c + 0], 0.0f);
        v.y = fmaxf(v.y * scale[c + 1] + shift[c + 1], 0.0f);
        v.z = fmaxf(v.z * scale[c + 2] + shift[c + 2], 0.0f);
        v.w = fmaxf(v.w * scale[c + 3] + shift[c + 3], 0.0f);
        *(float4*)(out + i) = v;
    }
}

// --------------------------------------------------------------------------
extern "C" void kernel_launch(void* const* d_in, const int* in_sizes, int n_in,
                              void* d_out, int out_size, void* d_ws, size_t ws_size,
                              hipStream_t stream) {
    (void)n_in; (void)out_size; (void)ws_size;
    const float* x    = (const float*)d_in[0];
    const int*   ei   = (const int*)  d_in[1];
    const float* eps  = (const float*)d_in[2];
    const float* W1   = (const float*)d_in[3];
    const float* b1   = (const float*)d_in[4];
    const float* g1   = (const float*)d_in[5];
    const float* be1  = (const float*)d_in[6];
    const float* W2   = (const float*)d_in[7];
    const float* b2   = (const float*)d_in[8];
    const float* g2   = (const float*)d_in[9];
    const float* be2  = (const float*)d_in[10];
    float* out = (float*)d_out;

    const int N = in_sizes[0] / IN_DIM;    // 100000
    const int E = in_sizes[1] / 2;         // 1600000

    // workspace layout: h1 [N*HID] then BN stats
    float* h1    = (float*)d_ws;
    float* stats = h1 + (size_t)N * HID;
    float* sum1 = stats;         float* sq1 = stats + 256;
    float* sc1  = stats + 512;   float* sh1 = stats + 768;
    float* sum2 = stats + 1024;  float* sq2 = stats + 1152;
    float* sc2  = stats + 1280;  float* sh2 = stats + 1408;
    const size_t statsN = 1536;

    float* aggr = out;  // reuse d_out as aggregation scratch (N*128 f32)

    const size_t aggrN = (size_t)N * IN_DIM;
    zero_kernel<<<(unsigned)((aggrN + 255) / 256), 256, 0, stream>>>(aggr, aggrN);
    zero_kernel<<<(unsigned)((statsN + 255) / 256), 256, 0, stream>>>(stats, statsN);

    scatter_kernel<<<2048, 256, 0, stream>>>(x, ei, aggr, E);

    const int mtiles = (N + 15) / 16;
    gemm1_kernel<<<dim3(mtiles, 2), 256, 0, stream>>>(x, aggr, eps, W1, b1, h1, N);

    colstats_kernel<HID><<<512, HID, 0, stream>>>(h1, N, sum1, sq1);
    bnprep_kernel<<<1, HID, 0, stream>>>(sum1, sq1, g1, be1, sc1, sh1, HID, 1.0f / (float)N);

    gemm2_kernel<<<dim3(mtiles), 256, 0, stream>>>(h1, sc1, sh1, W2, b2, out, N);

    colstats_kernel<OUT_DIM><<<512, OUT_DIM, 0, stream>>>(out, N, sum2, sq2);
    bnprep_kernel<<<1, OUT_DIM, 0, stream>>>(sum2, sq2, g2, be2, sc2, sh2, OUT_DIM, 1.0f / (float)N);

    const size_t outN = (size_t)N * OUT_DIM;
    bnrelu_kernel<<<(unsigned)((outN / 4 + 255) / 256), 256, 0, stream>>>(out, sc2, sh2, outN);
}